// RCD_Extractor_30872224923927
// MI455X (gfx1250) — compile-verified
//
#include <hip/hip_runtime.h>
#include <math.h>

typedef __attribute__((ext_vector_type(2))) float v2f;
typedef __attribute__((ext_vector_type(4))) float f4;
typedef __attribute__((ext_vector_type(8))) float v8f;

#define D_DIM 128
#define S_N   20000
#define E_N   10000
#define K_N   128
#define B_N   2048
#define NEK_N 40000
#define NSE_N 500000

// ---------------------------------------------------------------------------
// z = concat(h0[n0], h1) @ W   using V_WMMA_F32_16X16X4_F32.
// One wave per 16x16 output tile. grid = (n/16, 8), block = 32.
// A 16x4 f32 layout: lanes 0-15 K={0,1} in v0/v1, lanes 16-31 K={2,3}.
// C layout: VGPR i -> M=i (lanes 0-15) / M=8+i (lanes 16-31), N=lane&15.
// ---------------------------------------------------------------------------
__global__ void gemm_concat_wmma(const float* __restrict__ h0, int n0,
                                 const float* __restrict__ h1,
                                 const float* __restrict__ W,
                                 float* __restrict__ z) {
  const int mt   = blockIdx.x;
  const int nt   = blockIdx.y;
  const int lane = threadIdx.x;
  const int half = lane >> 4;
  const int l    = lane & 15;
  const int row  = mt * 16 + l;
  const float* __restrict__ hrow =
      (row < n0) ? (h0 + (size_t)row * D_DIM) : (h1 + (size_t)(row - n0) * D_DIM);
  const int col = nt * 16 + l;

  v8f acc = {};
  for (int k0 = 0; k0 < D_DIM; k0 += 4) {
    v2f a, b;
    a.x = hrow[k0 + 2 * half];
    a.y = hrow[k0 + 2 * half + 1];
    b.x = W[(size_t)(k0 + 2 * half) * D_DIM + col];
    b.y = W[(size_t)(k0 + 2 * half + 1) * D_DIM + col];
    acc = __builtin_amdgcn_wmma_f32_16x16x4_f32(false, a, false, b,
                                                (short)0, acc, false, false);
  }
  const int baseRow = mt * 16 + half * 8;
#pragma unroll
  for (int i = 0; i < 8; ++i)
    z[(size_t)(baseRow + i) * D_DIM + col] = acc[i];
}

// p[i] = z[i] . a[0:128] ; also init maxp = -inf, den = 0. One wave per row.
__global__ void node_dot(const float* __restrict__ z, const float* __restrict__ a,
                         float* __restrict__ p, float* __restrict__ maxp,
                         float* __restrict__ den, int n) {
  const int row  = (int)((blockIdx.x * blockDim.x + threadIdx.x) >> 5);
  const int lane = threadIdx.x & 31;
  if (row >= n) return;
  const f4 zv = ((const f4*)(z + (size_t)row * D_DIM))[lane];
  const f4 av = ((const f4*)a)[lane];
  float s = zv.x * av.x + zv.y * av.y + zv.z * av.z + zv.w * av.w;
  for (int o = 16; o; o >>= 1) s += __shfl_xor(s, o, 32);
  if (lane == 0) {
    p[row]    = s;
    maxp[row] = __int_as_float(0xFF800000);
    den[row]  = 0.0f;
  }
}

__device__ __forceinline__ void atomicMaxF32(float* addr, float v) {
  if (v >= 0.0f) atomicMax((int*)addr, __float_as_int(v));
  else           atomicMin((unsigned int*)addr, __float_as_uint(v));
}

// maxp[dst[j]] = max(maxp[dst[j]], p[src[j]]). One thread per edge.
__global__ void edge_max(const float* __restrict__ p, const int* __restrict__ src,
                         const int* __restrict__ dst, float* __restrict__ maxp, int nE) {
  const int j = (int)(blockIdx.x * blockDim.x + threadIdx.x);
  if (j >= nE) return;
  atomicMaxF32(maxp + dst[j], p[src[j]]);
}

// w = exp(p[src]-maxp[dst]); den[dst]+=w; out[dst] += w*z[src]. One wave/edge.
__global__ void edge_scatter(const float* __restrict__ z, const float* __restrict__ p,
                             const int* __restrict__ src, const int* __restrict__ dst,
                             const float* __restrict__ maxp, float* __restrict__ den,
                             float* __restrict__ out, int nE) {
  const int j    = (int)((blockIdx.x * blockDim.x + threadIdx.x) >> 5);
  const int lane = threadIdx.x & 31;
  if (j >= nE) return;
  const int s = src[j], d = dst[j];
  const float w = expf(p[s] - maxp[d]);
  if (lane == 0) atomicAdd(den + d, w);
  const f4 zv = ((const f4*)(z + (size_t)s * D_DIM))[lane];
  float* o = out + (size_t)d * D_DIM + lane * 4;
  atomicAdd(o + 0, w * zv.x);
  atomicAdd(o + 1, w * zv.y);
  atomicAdd(o + 2, w * zv.z);
  atomicAdd(o + 3, w * zv.w);
}

// out[i][:] /= (den[i] > 0 ? den[i] : 1)
__global__ void seg_norm(f4* __restrict__ out, const float* __restrict__ den, unsigned n4) {
  const unsigned t = blockIdx.x * blockDim.x + threadIdx.x;
  if (t >= n4) return;
  const float dn = den[t >> 5];
  const float sc = (dn > 0.0f) ? (1.0f / dn) : 1.0f;
  f4 v = out[t];
  v *= sc;
  out[t] = v;
}

// o = a + b (float4 granular)
__global__ void add_rows(const f4* __restrict__ a, const f4* __restrict__ b,
                         f4* __restrict__ o, unsigned n4) {
  const unsigned t = blockIdx.x * blockDim.x + threadIdx.x;
  if (t >= n4) return;
  o[t] = a[t] + b[t];
}

// ex2 = ex + w1*Bx + w2*Cx with [w1,w2] = softmax([ex,Bx]@fw0+fb0, [ex,Cx]@fw1+fb1)
// One wave per exercise row. Fw pre-offset to layer l (two rows of 256), Fb to Fb+l*3.
__global__ void combine_ex(const float* __restrict__ ex, const float* __restrict__ Bx,
                           const float* __restrict__ Cx, const float* __restrict__ Fw,
                           const float* __restrict__ Fb, float* __restrict__ out, int n) {
  const int row  = (int)((blockIdx.x * blockDim.x + threadIdx.x) >> 5);
  const int lane = threadIdx.x & 31;
  if (row >= n) return;
  const f4 e4  = ((const f4*)(ex + (size_t)row * D_DIM))[lane];
  const f4 b4  = ((const f4*)(Bx + (size_t)row * D_DIM))[lane];
  const f4 c4  = ((const f4*)(Cx + (size_t)row * D_DIM))[lane];
  const f4 f0a = ((const f4*)(Fw))[lane];
  const f4 f0b = ((const f4*)(Fw + 128))[lane];
  const f4 f1a = ((const f4*)(Fw + 256))[lane];
  const f4 f1b = ((const f4*)(Fw + 384))[lane];
  float s1 = e4.x * f0a.x + e4.y * f0a.y + e4.z * f0a.z + e4.w * f0a.w +
             b4.x * f0b.x + b4.y * f0b.y + b4.z * f0b.z + b4.w * f0b.w;
  float s2 = e4.x * f1a.x + e4.y * f1a.y + e4.z * f1a.z + e4.w * f1a.w +
             c4.x * f1b.x + c4.y * f1b.y + c4.z * f1b.z + c4.w * f1b.w;
  for (int o = 16; o; o >>= 1) {
    s1 += __shfl_xor(s1, o, 32);
    s2 += __shfl_xor(s2, o, 32);
  }
  s1 += Fb[0];
  s2 += Fb[1];
  const float m  = fmaxf(s1, s2);
  const float e1 = expf(s1 - m), e2 = expf(s2 - m);
  const float inv = 1.0f / (e1 + e2);
  const float w1 = e1 * inv, w2 = e2 * inv;
  f4 o4;
  o4.x = e4.x + w1 * b4.x + w2 * c4.x;
  o4.y = e4.y + w1 * b4.y + w2 * c4.y;
  o4.z = e4.z + w1 * b4.z + w2 * c4.z;
  o4.w = e4.w + w1 * b4.w + w2 * c4.w;
  ((f4*)(out + (size_t)row * D_DIM))[lane] = o4;
}

// out[b,i,:] = tab[ids[b],:] for i in 0..127. float4 granular, nontemporal stores.
__global__ void bcast_rows(const float* __restrict__ tab, const int* __restrict__ ids,
                           f4* __restrict__ out, unsigned total4) {
  const unsigned t = blockIdx.x * blockDim.x + threadIdx.x;
  if (t >= total4) return;
  const unsigned j = t & 31;          // float4 col within row (128/4)
  const unsigned b = t >> 12;         // 32 * 128 float4 per batch item
  const f4 v = ((const f4*)tab)[(size_t)ids[b] * 32 + j];
  __builtin_nontemporal_store(v, out + t);
}

// out[b,k,:] = kn[k,:] broadcast across batch
__global__ void kn_bcast(const float* __restrict__ kn, f4* __restrict__ out,
                         unsigned total4) {
  const unsigned t = blockIdx.x * blockDim.x + threadIdx.x;
  if (t >= total4) return;
  const f4 v = ((const f4*)kn)[t & 4095];  // K*D/4 = 4096
  __builtin_nontemporal_store(v, out + t);
}

__global__ void disc_gather(const float* __restrict__ disc, const int* __restrict__ eid,
                            float* __restrict__ out, int n) {
  const int b = (int)(blockIdx.x * blockDim.x + threadIdx.x);
  if (b >= n) return;
  out[b] = disc[eid[b]];
}

// ---------------------------------------------------------------------------
static void graph_layer(const float* h0, int n0, const float* h1, int n1,
                        const float* W, const float* aVec,
                        const int* src, const int* dst, int nE,
                        float* z, float* p, float* maxp, float* den,
                        float* out, hipStream_t stream) {
  const int n = n0 + n1;
  hipMemsetAsync(out, 0, (size_t)n * D_DIM * sizeof(float), stream);
  dim3 g(n / 16, D_DIM / 16);
  gemm_concat_wmma<<<g, 32, 0, stream>>>(h0, n0, h1, W, z);
  node_dot<<<(n + 7) / 8, 256, 0, stream>>>(z, aVec, p, maxp, den, n);
  edge_max<<<(nE + 255) / 256, 256, 0, stream>>>(p, src, dst, maxp, nE);
  edge_scatter<<<(nE + 7) / 8, 256, 0, stream>>>(z, p, src, dst, maxp, den, out, nE);
  const unsigned n4 = (unsigned)n * (D_DIM / 4);
  seg_norm<<<(n4 + 255) / 256, 256, 0, stream>>>((f4*)out, den, n4);
}

extern "C" void kernel_launch(void* const* d_in, const int* in_sizes, int n_in,
                              void* d_out, int out_size, void* d_ws, size_t ws_size,
                              hipStream_t stream) {
  const float* stu  = (const float*)d_in[0];
  const float* exe  = (const float*)d_in[1];
  const float* kn   = (const float*)d_in[2];
  const float* disc = (const float*)d_in[3];
  const float* GW   = (const float*)d_in[4];
  const float* Ga   = (const float*)d_in[5];
  const float* Fw   = (const float*)d_in[6];
  const float* Fb   = (const float*)d_in[7];
  const int* sid    = (const int*)d_in[9];
  const int* eid    = (const int*)d_in[10];
  const int* ek_src = (const int*)d_in[11];
  const int* ek_dst = (const int*)d_in[12];
  const int* se_src = (const int*)d_in[13];
  const int* se_dst = (const int*)d_in[14];

  // Workspace layout (floats). Total ~21.9M floats (~84 MB).
  float* ws = (float*)d_ws;
  float* Z    = ws;                   // 30000*128
  float* P    = Z + 3840000;          // 30000
  float* MAXP = P + 30000;            // 30000
  float* DEN  = MAXP + 30000;         // 30000
  float* KFE  = DEN + 30000;          // 10128*128
  float* EFK  = KFE + 1296384;        // 10128*128
  float* UFE  = EFK + 1296384;        // 30000*128
  float* EFU  = UFE + 3840000;        // 30000*128
  float* KNA  = EFU + 3840000;        // 128*128
  float* EXA  = KNA + 16384;          // 10000*128
  float* STA  = EXA + 1280000;        // 20000*128
  float* KNB  = STA + 2560000;
  float* EXB  = KNB + 16384;
  float* STB  = EXB + 1280000;

  for (int l = 0; l < 2; ++l) {
    const float* kin = (l == 0) ? kn  : KNA;
    const float* ein = (l == 0) ? exe : EXA;
    const float* tin = (l == 0) ? stu : STA;
    float* kout = (l == 0) ? KNA : KNB;
    float* eout = (l == 0) ? EXA : EXB;
    float* tout = (l == 0) ? STA : STB;
    const float* GWl = GW + (size_t)l * 4 * D_DIM * D_DIM;
    const float* Gal = Ga + (size_t)l * 4 * 2 * D_DIM;

    // kfe: exercises+knowledge, edges ek_src -> ek_dst
    graph_layer(ein, E_N, kin, K_N, GWl + 0 * D_DIM * D_DIM, Gal + 0 * 2 * D_DIM,
                ek_src, ek_dst, NEK_N, Z, P, MAXP, DEN, KFE, stream);
    // efk: reversed edges
    graph_layer(ein, E_N, kin, K_N, GWl + 1 * D_DIM * D_DIM, Gal + 1 * 2 * D_DIM,
                ek_dst, ek_src, NEK_N, Z, P, MAXP, DEN, EFK, stream);
    // ufe: exercises+students, edges se_src -> se_dst
    graph_layer(ein, E_N, tin, S_N, GWl + 2 * D_DIM * D_DIM, Gal + 2 * 2 * D_DIM,
                se_src, se_dst, NSE_N, Z, P, MAXP, DEN, UFE, stream);
    // efu: reversed edges
    graph_layer(ein, E_N, tin, S_N, GWl + 3 * D_DIM * D_DIM, Gal + 3 * 2 * D_DIM,
                se_dst, se_src, NSE_N, Z, P, MAXP, DEN, EFU, stream);

    // kn2 = kn + kfe[E:]   (singleton-axis softmax == 1)
    {
      const unsigned n4 = K_N * (D_DIM / 4);
      add_rows<<<(n4 + 255) / 256, 256, 0, stream>>>(
          (const f4*)kin, (const f4*)(KFE + (size_t)E_N * D_DIM), (f4*)kout, n4);
    }
    // st2 = st + ufe[E:]
    {
      const unsigned n4 = (unsigned)S_N * (D_DIM / 4);
      add_rows<<<(n4 + 255) / 256, 256, 0, stream>>>(
          (const f4*)tin, (const f4*)(UFE + (size_t)E_N * D_DIM), (f4*)tout, n4);
    }
    // ex2 via 2-way softmax fusion with Bx=efk[:E], Cx=efu[:E]
    combine_ex<<<(E_N + 7) / 8, 256, 0, stream>>>(
        ein, EFK, EFU, Fw + (size_t)l * 3 * 2 * D_DIM, Fb + (size_t)l * 3, eout, E_N);
  }

  // Outputs (flat, in return order): stu_ts, exer_ts, disc, kn_ts
  float* out = (float*)d_out;
  float* o_stu  = out;                                   // B*D*D
  float* o_exer = o_stu + (size_t)B_N * D_DIM * D_DIM;   // B*D*D
  float* o_disc = o_exer + (size_t)B_N * D_DIM * D_DIM;  // B
  float* o_kn   = o_disc + B_N;                          // B*K*D

  const unsigned bc4 = (unsigned)B_N * D_DIM * (D_DIM / 4);  // 8388608
  bcast_rows<<<(bc4 + 255) / 256, 256, 0, stream>>>(STB, sid, (f4*)o_stu, bc4);
  bcast_rows<<<(bc4 + 255) / 256, 256, 0, stream>>>(EXB, eid, (f4*)o_exer, bc4);
  disc_gather<<<(B_N + 255) / 256, 256, 0, stream>>>(disc, eid, o_disc, B_N);
  const unsigned kn4 = (unsigned)B_N * K_N * (D_DIM / 4);
  kn_bcast<<<(kn4 + 255) / 256, 256, 0, stream>>>(KNB, (f4*)o_kn, kn4);
}